// MultiscaleMessagePassing_34273839022284
// MI455X (gfx1250) — compile-verified
//
#include <hip/hip_runtime.h>

typedef __attribute__((ext_vector_type(2))) float v2f;
typedef __attribute__((ext_vector_type(4))) float v4f;
typedef __attribute__((ext_vector_type(8))) float v8f;

#define HID 128
#define KSTEPS (HID / 4)   // 32 k-steps of 4 for f32 WMMA
#define NTILES (HID / 16)  // 8 tiles of 16 output cols
#define WAVES_PER_BLOCK 4

__device__ __forceinline__ void fatomic_add(float* p, float v) {
  __hip_atomic_fetch_add(p, v, __ATOMIC_RELAXED, __HIP_MEMORY_SCOPE_AGENT);
}

// ---------------- zero output + counts (non-temporal stores) ----------------
__global__ void zero_kernel(float* __restrict__ out, long out_n,
                            float* __restrict__ cnt, int cnt_n) {
  long stride = (long)gridDim.x * blockDim.x;
  long total = out_n + cnt_n;
  for (long i = (long)blockIdx.x * blockDim.x + threadIdx.x; i < total; i += stride) {
    if (i < out_n) __builtin_nontemporal_store(0.0f, out + i);
    else cnt[i - out_n] = 0.0f;
  }
}

// ---------------- segment counts (nodes + edges) ----------------
__global__ void count_kernel(const int* __restrict__ cent, long N,
                             const int* __restrict__ eseg, long E,
                             const int* __restrict__ p_nc,
                             const int* __restrict__ p_nce,
                             float* __restrict__ cnt) {
  const int nc  = *p_nc;
  const int nce = *p_nce;
  long stride = (long)gridDim.x * blockDim.x;
  long total = N + E;
  for (long i = (long)blockIdx.x * blockDim.x + threadIdx.x; i < total; i += stride) {
    if (i < N) {
      fatomic_add(&cnt[cent[i]], 1.0f);
    } else {
      int s = eseg[i - N];
      if (s < nce) fatomic_add(&cnt[nc + s], 1.0f);
    }
  }
}

// ------- fused MLP (fp32 WMMA) + pre-scaled node scatter (mean on the fly) -------
__global__ void __launch_bounds__(32 * WAVES_PER_BLOCK)
mlp_scatter_kernel(const float* __restrict__ x, const float* __restrict__ dist,
                   const float* __restrict__ W1, const float* __restrict__ b1,
                   const float* __restrict__ W2, const float* __restrict__ b2,
                   const int* __restrict__ cent, const float* __restrict__ cnt,
                   float* __restrict__ out, int N, int total_segs) {
  // per-wave activation staging tile, padded to 129 floats/row (conflict-free)
  __shared__ float act[WAVES_PER_BLOCK][16][HID + 1];

  const int lane = threadIdx.x & 31;
  const int wv   = threadIdx.x >> 5;
  const int hi   = lane >> 4;   // half-wave select (K pair / row+8)
  const int lo   = lane & 15;   // row / col within 16
  const int numTiles = (N + 15) >> 4;
  const int tile = blockIdx.x * WAVES_PER_BLOCK + wv;
  if (tile >= numTiles) return;
  const int node0 = tile * 16;
  const int batch = blockIdx.y;

  // ---- GEMM1: [16 x 128] = concat(x,d)[16 x 129] @ W1[129 x 128] ----
  int arow = node0 + lo; if (arow >= N) arow = N - 1;
  const float* xr = x + ((size_t)batch * N + arow) * HID;

  v8f acc[NTILES];
#pragma unroll
  for (int nt = 0; nt < NTILES; ++nt) { v8f z = {}; acc[nt] = z; }

  for (int ks = 0; ks < KSTEPS; ++ks) {
    const int k = ks * 4 + 2 * hi;              // A 16x4 layout: lane hi selects K pair
    v2f a;
    a.x = __builtin_nontemporal_load(xr + k);   // x is streamed exactly once
    a.y = __builtin_nontemporal_load(xr + k + 1);
#pragma unroll
    for (int nt = 0; nt < NTILES; ++nt) {
      const int n = nt * 16 + lo;               // B 4x16 layout: lane lo = column
      v2f bf;
      bf.x = W1[(size_t)k * HID + n];
      bf.y = W1[(size_t)(k + 1) * HID + n];
      acc[nt] = __builtin_amdgcn_wmma_f32_16x16x4_f32(
          false, a, false, bf, (short)0, acc[nt], false, false);
    }
  }

  // rank-1 update for the distance column, + bias, SiLU, stage to LDS
#pragma unroll
  for (int nt = 0; nt < NTILES; ++nt) {
    const int n = nt * 16 + lo;
    const float wl = W1[(size_t)HID * HID + n]; // W1 row 128 (distance row)
    const float bb = b1[n];
#pragma unroll
    for (int r = 0; r < 8; ++r) {
      int node = node0 + r + 8 * hi;            // C layout: M = r + 8*hi, N = lo
      int nq = node < N ? node : N - 1;
      float t = acc[nt][r] + dist[nq] * wl + bb;
      float s = t / (1.0f + __expf(-t));        // SiLU
      act[wv][r + 8 * hi][n] = s;
    }
  }
  asm volatile("s_wait_dscnt 0" ::: "memory");  // same-wave LDS RAW

  // ---- GEMM2: [16 x 128] = act[16 x 128] @ W2[128 x 128] ----
  v8f acc2[NTILES];
#pragma unroll
  for (int nt = 0; nt < NTILES; ++nt) { v8f z = {}; acc2[nt] = z; }

  for (int ks = 0; ks < KSTEPS; ++ks) {
    const int k = ks * 4 + 2 * hi;
    v2f a; a.x = act[wv][lo][k]; a.y = act[wv][lo][k + 1];
#pragma unroll
    for (int nt = 0; nt < NTILES; ++nt) {
      const int n = nt * 16 + lo;
      v2f bf;
      bf.x = W2[(size_t)k * HID + n];
      bf.y = W2[(size_t)(k + 1) * HID + n];
      acc2[nt] = __builtin_amdgcn_wmma_f32_16x16x4_f32(
          false, a, false, bf, (short)0, acc2[nt], false, false);
    }
  }

  // bias + atomic scatter-add of (value / segment_count) -> mean accumulates directly
#pragma unroll
  for (int r = 0; r < 8; ++r) {
    const int node = node0 + r + 8 * hi;
    if (node >= N) continue;
    const int cid = cent[node];
    const float inv = 1.0f / fmaxf(cnt[cid], 1.0f);
    float* orow = out + ((size_t)batch * total_segs + cid) * HID;
#pragma unroll
    for (int nt = 0; nt < NTILES; ++nt) {
      const int n = nt * 16 + lo;
      fatomic_add(&orow[n], (acc2[nt][r] + b2[n]) * inv);
    }
  }
}

// -------- edge scatter: one wave per edge, pre-scaled by 1/count --------
__global__ void __launch_bounds__(256)
edge_scatter_kernel(const float* __restrict__ ea, const int* __restrict__ eseg,
                    const int* __restrict__ p_nc, const int* __restrict__ p_nce,
                    const float* __restrict__ cnt, float* __restrict__ out,
                    long E, int total_segs) {
  const int lane = threadIdx.x & 31;
  const int wv   = threadIdx.x >> 5;
  const long e = (long)blockIdx.x * 8 + wv;
  if (e >= E) return;
  const int seg = eseg[e];
  const int nce = *p_nce;
  if (seg >= nce) return;                       // dump bucket: intra-voxel edges
  const int nc = *p_nc;
  const int batch = blockIdx.y;
  const float inv = 1.0f / fmaxf(cnt[nc + seg], 1.0f);
  // edge_attr is streamed exactly once: non-temporal b128 load (clang ext-vector)
  const v4f* src = (const v4f*)(ea + ((size_t)batch * E + e) * HID + lane * 4);
  const v4f v = __builtin_nontemporal_load(src);
  float* dst = out + ((size_t)batch * total_segs + nc + seg) * HID + lane * 4;
  fatomic_add(dst + 0, v.x * inv);
  fatomic_add(dst + 1, v.y * inv);
  fatomic_add(dst + 2, v.z * inv);
  fatomic_add(dst + 3, v.w * inv);
}

extern "C" void kernel_launch(void* const* d_in, const int* in_sizes, int n_in,
                              void* d_out, int out_size, void* d_ws, size_t ws_size,
                              hipStream_t stream) {
  const float* x    = (const float*)d_in[0];
  const float* dist = (const float*)d_in[1];
  const float* ea   = (const float*)d_in[2];
  const float* W1   = (const float*)d_in[3];
  const float* b1   = (const float*)d_in[4];
  const float* W2   = (const float*)d_in[5];
  const float* b2   = (const float*)d_in[6];
  const int* cent   = (const int*)d_in[7];
  const int* eseg   = (const int*)d_in[8];
  const int* p_nc   = (const int*)d_in[9];   // n_coarse (device scalar)
  const int* p_nce  = (const int*)d_in[10];  // n_coarse_edges (device scalar)

  const long H = in_sizes[4];                // 128
  const long N = in_sizes[1];                // distances has N elements
  const long B = in_sizes[0] / (N * H);
  const long E = in_sizes[2] / (B * H);
  const long out_n = (long)out_size;
  const int  total_segs = (int)(out_n / (B * H));  // n_coarse + n_coarse_edges

  float* out = (float*)d_out;
  float* cnt = (float*)d_ws;                 // total_segs floats of scratch

  {
    long tot = out_n + total_segs;
    long blocks = (tot + 1023) / 1024; if (blocks > 1048576) blocks = 1048576;
    zero_kernel<<<dim3((unsigned)blocks), 256, 0, stream>>>(out, out_n, cnt, total_segs);
  }
  {
    long tot = N + E;
    long blocks = (tot + 255) / 256; if (blocks > 262144) blocks = 262144;
    count_kernel<<<dim3((unsigned)blocks), 256, 0, stream>>>(cent, N, eseg, E, p_nc, p_nce, cnt);
  }
  {
    int numTiles = (int)((N + 15) / 16);
    dim3 grid((numTiles + WAVES_PER_BLOCK - 1) / WAVES_PER_BLOCK, (unsigned)B);
    mlp_scatter_kernel<<<grid, 32 * WAVES_PER_BLOCK, 0, stream>>>(
        x, dist, W1, b1, W2, b2, cent, cnt, out, (int)N, total_segs);
  }
  {
    dim3 grid((unsigned)((E + 7) / 8), (unsigned)B);
    edge_scatter_kernel<<<grid, 256, 0, stream>>>(ea, eseg, p_nc, p_nce, cnt, out, E, total_segs);
  }
}